// _GCNEncoder_83296595739286
// MI455X (gfx1250) — compile-verified
//
#include <hip/hip_runtime.h>
#include <hip/hip_fp16.h>

typedef float fv2 __attribute__((ext_vector_type(2)));
typedef float fv8 __attribute__((ext_vector_type(8)));

#define HID_DIM 256

__device__ __forceinline__ void wait_async_zero() {
#if __has_builtin(__builtin_amdgcn_s_wait_asynccnt)
  __builtin_amdgcn_s_wait_asynccnt(0);
#else
  asm volatile("s_wait_asynccnt 0" ::: "memory");
#endif
}

// CDNA5 async global->LDS DMA (ASYNCcnt path), 16 bytes per lane.
// VDST = LDS byte address (32-bit VGPR), VADDR = 64-bit global address, no SADDR.
__device__ __forceinline__ void async_load_lds_b128(void* lds_generic,
                                                    const void* global_generic) {
  unsigned lds_off = (unsigned)(unsigned long long)(
      __attribute__((address_space(3))) void*)lds_generic;
  unsigned long long gaddr = (unsigned long long)global_generic;
  asm volatile("global_load_async_to_lds_b128 %0, %1, off"
               :: "v"(lds_off), "v"(gaddr)
               : "memory");
}

// ---------------- degree / normalization ----------------

__global__ __launch_bounds__(256) void k_init_deg(float* __restrict__ deg, int n) {
  int i = blockIdx.x * 256 + threadIdx.x;
  if (i < n) deg[i] = 1.0f;  // self-loop contributes 1 to every row-degree
}

__global__ __launch_bounds__(256) void k_edge_deg(const int* __restrict__ s,
                                                  const int* __restrict__ d,
                                                  float* __restrict__ deg, int E) {
  int e = blockIdx.x * 256 + threadIdx.x;
  if (e < E) {
    unsafeAtomicAdd(&deg[s[e]], 1.0f);  // forward edge: row = src
    unsafeAtomicAdd(&deg[d[e]], 1.0f);  // reversed edge: row = dst
  }
}

__global__ __launch_bounds__(256) void k_rsqrt(float* __restrict__ deg, int n) {
  int i = blockIdx.x * 256 + threadIdx.x;
  if (i < n) deg[i] = 1.0f / sqrtf(deg[i]);  // deg >= 1 always
}

// ---------------- aggregation ----------------
// init: agg[i] = h[i] * dis[i]^2   (self-loop term, also initializes buffer)
__global__ __launch_bounds__(256) void k_init_agg(const float* __restrict__ h,
                                                  const float* __restrict__ dis,
                                                  float* __restrict__ agg,
                                                  unsigned n, int cshift) {
  unsigned t = blockIdx.x * 256u + threadIdx.x;
  unsigned total = n << cshift;
  if (t >= total) return;
  unsigned i = t >> cshift;
  unsigned c = t & ((1u << cshift) - 1u);
  float ds = dis[i];
  float dsq = ds * ds;
  size_t ofs = ((size_t)i << (cshift + 2)) + (size_t)c * 4;
  float4 v = *(const float4*)(h + ofs);
  float4 o; o.x = v.x * dsq; o.y = v.y * dsq; o.z = v.z * dsq; o.w = v.w * dsq;
  *(float4*)(agg + ofs) = o;
}

// Each thread handles one float4 chunk of one edge, both directions:
//   agg[src] += h[dst]*norm ; agg[dst] += h[src]*norm
__global__ __launch_bounds__(256) void k_aggregate(const float* __restrict__ h,
                                                   float* __restrict__ agg,
                                                   const float* __restrict__ dis,
                                                   const int* __restrict__ src,
                                                   const int* __restrict__ dst,
                                                   unsigned E, int cshift) {
  unsigned t = blockIdx.x * 256u + threadIdx.x;
  unsigned total = E << cshift;
  if (t >= total) return;
  unsigned e = t >> cshift;
  unsigned c = t & ((1u << cshift) - 1u);
  int s = src[e], d = dst[e];
  float nrm = dis[s] * dis[d];
  size_t so = ((size_t)s << (cshift + 2)) + (size_t)c * 4;
  size_t dofs = ((size_t)d << (cshift + 2)) + (size_t)c * 4;
  float4 hd = *(const float4*)(h + dofs);
  float4 hs = *(const float4*)(h + so);
  float* as = agg + so;
  float* ad = agg + dofs;
  unsafeAtomicAdd(as + 0, hd.x * nrm);
  unsafeAtomicAdd(as + 1, hd.y * nrm);
  unsafeAtomicAdd(as + 2, hd.z * nrm);
  unsafeAtomicAdd(as + 3, hd.w * nrm);
  unsafeAtomicAdd(ad + 0, hs.x * nrm);
  unsafeAtomicAdd(ad + 1, hs.y * nrm);
  unsafeAtomicAdd(ad + 2, hs.z * nrm);
  unsafeAtomicAdd(ad + 3, hs.w * nrm);
}

// ---------------- fp32 WMMA GEMM:  out[N x 256] = A[N x K] * W^T + bias ----------------
// W is [HID x K] row-major, so B[k][n] = W[n*K + k].
// One wave per 16x16 output tile; 8 waves/block share one 16xK A-tile in LDS,
// staged via CDNA5 async global->LDS DMA (ASYNCcnt).
template <int KDIM, bool RELU>
__global__ __launch_bounds__(256) void k_gemm_wmma(const float* __restrict__ A,
                                                   const float* __restrict__ W,
                                                   const float* __restrict__ bias,
                                                   float* __restrict__ out,
                                                   int nrows) {
  __shared__ float As[16][KDIM + 4];  // +4 pad: 16-lane column reads hit distinct banks
  const int tid = threadIdx.x;
  const int mtile = blockIdx.x * 16;
  constexpr int C4 = KDIM / 4;

  for (int i = tid; i < 16 * C4; i += 256) {
    int r = i / C4, c = i % C4;
    int row = mtile + r;
    if (row < nrows) {
      async_load_lds_b128(&As[r][c * 4], A + (size_t)row * KDIM + (size_t)c * 4);
    } else {
      *(float4*)(&As[r][c * 4]) = make_float4(0.f, 0.f, 0.f, 0.f);
    }
  }
  wait_async_zero();
  __syncthreads();

  const int wave = tid >> 5;
  const int lane = tid & 31;
  const int half = lane >> 4;     // lane half selects K sub-pair (ISA A/B layout)
  const int lane16 = lane & 15;
  const int ntile = (blockIdx.y * 8 + wave) * 16;

  const float* arow = &As[lane16][half * 2];
  const float* wrow = W + (size_t)(ntile + lane16) * KDIM + half * 2;

  fv8 acc = {0.f, 0.f, 0.f, 0.f, 0.f, 0.f, 0.f, 0.f};
#pragma unroll 4
  for (int k0 = 0; k0 < KDIM; k0 += 4) {
    fv2 a, b;
    a.x = arow[k0];     // A[M=lane16][k0 + half*2]
    a.y = arow[k0 + 1];
    b.x = wrow[k0];     // B[k0 + half*2][N=lane16] = W[(ntile+lane16)*K + k]
    b.y = wrow[k0 + 1];
    acc = __builtin_amdgcn_wmma_f32_16x16x4_f32(false, a, false, b, (short)0, acc,
                                                false, false);
  }

  const float bv = bias[ntile + lane16];
  float* outp = out + (size_t)(mtile + half * 8) * HID_DIM + ntile + lane16;
  if (mtile + 16 <= nrows) {
    // full tile: straight-line stores, no per-row predication
#pragma unroll
    for (int v = 0; v < 8; v++) {
      float val = acc[v] + bv;
      if (RELU) val = fmaxf(val, 0.0f);
      outp[(size_t)v * HID_DIM] = val;
    }
  } else {
#pragma unroll
    for (int v = 0; v < 8; v++) {
      int row = mtile + v + half * 8;
      float val = acc[v] + bv;
      if (RELU) val = fmaxf(val, 0.0f);
      if (row < nrows) out[(size_t)row * HID_DIM + ntile + lane16] = val;
    }
  }
}

// ---------------- launch ----------------

extern "C" void kernel_launch(void* const* d_in, const int* in_sizes, int n_in,
                              void* d_out, int out_size, void* d_ws, size_t ws_size,
                              hipStream_t stream) {
  const float* x  = (const float*)d_in[0];   // N x 128
  const float* W1 = (const float*)d_in[1];   // 256 x 128
  const float* b1 = (const float*)d_in[2];   // 256
  const float* W2 = (const float*)d_in[3];   // 256 x 256
  const float* b2 = (const float*)d_in[4];   // 256
  const int*   ei = (const int*)d_in[5];     // 2 x E
  const int N = in_sizes[0] / 128;
  const int E = in_sizes[5] / 2;
  const int* ei_src = ei;
  const int* ei_dst = ei + E;
  float* out = (float*)d_out;

  // workspace: dis | agg (N x 256 region, reused as N x 128 for layer 1) | h1 (N x 256)
  char* ws = (char*)d_ws;
  float* dis = (float*)ws;
  size_t off = (((size_t)N * 4) + 255) & ~(size_t)255;
  float* agg = (float*)(ws + off);
  off += (size_t)N * HID_DIM * 4;
  float* h1 = (float*)(ws + off);

  const int nb_n = (N + 255) / 256;
  const int nb_e = (E + 255) / 256;

  // degrees -> dis = deg^-1/2
  k_init_deg<<<nb_n, 256, 0, stream>>>(dis, N);
  k_edge_deg<<<nb_e, 256, 0, stream>>>(ei_src, ei_dst, dis, E);
  k_rsqrt<<<nb_n, 256, 0, stream>>>(dis, N);

  const int mtiles = (N + 15) / 16;

  // ---- layer 1: dim = 128 (cshift = 5 -> 32 float4 chunks/row) ----
  {
    unsigned tot_i = (unsigned)N << 5;
    k_init_agg<<<(tot_i + 255) / 256, 256, 0, stream>>>(x, dis, agg, (unsigned)N, 5);
    unsigned tot_e = (unsigned)E << 5;
    k_aggregate<<<(tot_e + 255) / 256, 256, 0, stream>>>(x, agg, dis, ei_src, ei_dst,
                                                         (unsigned)E, 5);
    dim3 g(mtiles, HID_DIM / 16 / 8);
    k_gemm_wmma<128, true><<<g, 256, 0, stream>>>(agg, W1, b1, h1, N);
  }

  // ---- layer 2: dim = 256 (cshift = 6 -> 64 float4 chunks/row) ----
  {
    unsigned tot_i = (unsigned)N << 6;
    k_init_agg<<<(tot_i + 255) / 256, 256, 0, stream>>>(h1, dis, agg, (unsigned)N, 6);
    unsigned tot_e = (unsigned)E << 6;
    k_aggregate<<<(tot_e + 255) / 256, 256, 0, stream>>>(h1, agg, dis, ei_src, ei_dst,
                                                         (unsigned)E, 6);
    dim3 g(mtiles, HID_DIM / 16 / 8);
    k_gemm_wmma<256, false><<<g, 256, 0, stream>>>(agg, W2, b2, out, N);
  }
}